// CPE_35845797052436
// MI455X (gfx1250) — compile-verified
//
#include <hip/hip_runtime.h>

typedef float v2f __attribute__((ext_vector_type(2)));
typedef float v8f __attribute__((ext_vector_type(8)));

#define KWIN   9
#define CCH    256
#define GH     8
#define HD     32
#define PADW   4
#define TLEN   8192
#define BATCH  4
#define MASKV  (-1.0e9f)

__device__ __forceinline__ v8f splat8(float v) {
    return (v8f){v, v, v, v, v, v, v, v};
}

__device__ __forceinline__ v8f wmma_f32_4(v2f a, v2f b, v8f c) {
    // D = A(16x4,f32) * B(4x16,f32) + C(16x16,f32)
    return __builtin_amdgcn_wmma_f32_16x16x4_f32(
        /*neg_a=*/false, a, /*neg_b=*/false, b,
        /*c_mod=*/(short)0, c, /*reuse_a=*/false, /*reuse_b=*/false);
}

// Async global->LDS copy (CDNA5, tracked by ASYNCcnt). LDS address is the low
// 32 bits of the generic pointer.
__device__ __forceinline__ void async_copy_b128(const float* gptr, const float* lptr) {
    unsigned lds_off = (unsigned)(size_t)lptr;
    unsigned long long ga = (unsigned long long)(size_t)gptr;
    asm volatile("global_load_async_to_lds_b128 %0, %1, off"
                 :: "v"(lds_off), "v"(ga) : "memory");
}

// ---------------------------------------------------------------------------
// Kernel D: pair-interleave the V-half of kv_w once per call:
//   w_il[q*512 + c*2 + j] = kv_w[(2q+j)*512 + 256 + c],  q<128, c<256, j<2
// This makes every WMMA B operand (weight rows 2q, 2q+1 at one column) a
// contiguous 8-byte pair, so kernel A reads it with a single ds_load_b64.
// ---------------------------------------------------------------------------
__global__ __launch_bounds__(256) void interleave_w_kernel(
    const float* __restrict__ kv_w,   // (C, 2C)
    float* __restrict__ w_il)         // (128, 256, 2)
{
    const int tid = blockIdx.x * 256 + threadIdx.x;   // 32768 threads
    const int q = tid >> 8;
    const int c = tid & 255;
    v2f v;
    v.x = kv_w[(size_t)(2 * q)     * (2 * CCH) + CCH + c];
    v.y = kv_w[(size_t)(2 * q + 1) * (2 * CCH) + CCH + c];
    *(v2f*)&w_il[(size_t)q * 512 + c * 2] = v;
}

// ---------------------------------------------------------------------------
// Kernel A: y_v[b, p, j] = sum_c x[b, c, p] * kv_w[c, 256+j] + kv_b[256+j]
// One wave computes a 16-position x 256-channel strip (64 K-steps x 16 tiles
// of V_WMMA_F32_16X16X4_F32). The interleaved weights are staged per-block
// into LDS via async global->LDS DMA, double-buffered in 4-pair (8-row)
// slabs. Pair-row stride 544 floats: a b64 read touches banks {2m, 2m+1};
// the lane-halves read pair-rows 1 apart -> bank delta 544 % 64 == 32, so
// the two halves hit disjoint 32-bank sets (conflict-free).
// ---------------------------------------------------------------------------
#define PSTR   544                 // floats per pair-row in LDS (512 + pad)
#define SLABP  4                   // pair-rows per slab (= 8 k-rows, 2 K-steps)
#define NSLAB  (CCH / (2 * SLABP)) // 32 slabs
#define BUFSZ  (SLABP * PSTR)      // floats per slab buffer (2176)

__global__ __launch_bounds__(256) void kv_gemm_kernel(
    const float* __restrict__ x,     // (B, C, T)
    const float* __restrict__ w_il,  // (128, 256, 2) interleaved V weights
    const float* __restrict__ kv_b,  // (2C,)
    float* __restrict__ y_v)         // (B, T, C)
{
    __shared__ float wbuf[2 * BUFSZ];   // 17 KB, double buffered

    const int tid  = threadIdx.x;
    const int lane = tid & 31;
    const int wave = tid >> 5;
    const int sid  = blockIdx.x * 8 + wave;        // 2048 strips total
    const int bb   = sid >> 9;                     // / (TLEN/16)
    const int t0   = (sid & 511) << 4;
    const int m    = lane & 15;
    const int hi   = lane >> 4;

    // copy roles: 256 threads move 4 pair-rows x 512 floats per slab
    const int cq   = tid >> 6;          // 0..3 (pair-row within slab)
    const int cseg = (tid & 63) * 8;    // 8-float segment (0..504)

    v8f acc[16];
#pragma unroll
    for (int tile = 0; tile < 16; ++tile) {
        acc[tile] = splat8(kv_b[CCH + tile * 16 + m]);
    }

    const float* xb = x + (size_t)bb * CCH * TLEN + t0 + m;

    // prologue: slab 0
    {
        const float* g = w_il + (size_t)cq * 512 + cseg;
        const float* l = &wbuf[cq * PSTR + cseg];
        async_copy_b128(g,     l);
        async_copy_b128(g + 4, l + 4);
    }

    for (int s = 0; s < NSLAB; ++s) {
        if (s + 1 < NSLAB) {
            // prefetch next slab into the other buffer (freed by the barrier
            // at the end of iteration s-1)
            const float* g = w_il + (size_t)((s + 1) * SLABP + cq) * 512 + cseg;
            const float* l = &wbuf[((s + 1) & 1) * BUFSZ + cq * PSTR + cseg];
            async_copy_b128(g,     l);
            async_copy_b128(g + 4, l + 4);
            // warm the x stream for the next slab (2 rows per lane via hi)
            __builtin_prefetch(&xb[(size_t)((s + 1) * 8 + 2 * hi) * TLEN], 0, 1);
            __builtin_prefetch(&xb[(size_t)((s + 1) * 8 + 4 + 2 * hi) * TLEN], 0, 1);
            // async loads complete in order: first 2 (slab s) done when cnt<=2
            asm volatile("s_wait_asynccnt 0x2" ::: "memory");
        } else {
            asm volatile("s_wait_asynccnt 0x0" ::: "memory");
        }
        __syncthreads();

        const float* wsl = &wbuf[(s & 1) * BUFSZ];
        const int kbase = s * 8;
#pragma unroll
        for (int ks = 0; ks < 2; ++ks) {
            const int kk = kbase + ks * 4 + 2 * hi;
            v2f a;
            a.x = xb[(size_t)kk * TLEN];
            a.y = xb[(size_t)(kk + 1) * TLEN];
            // this half-wave's pair-row: weight rows (kk, kk+1) interleaved
            const v2f* wp = (const v2f*)(wsl + (2 * ks + hi) * PSTR);
#pragma unroll
            for (int tile = 0; tile < 16; ++tile) {
                const v2f bm = wp[tile * 16 + m];   // one ds_load_b64
                acc[tile] = wmma_f32_4(a, bm, acc[tile]);
            }
        }
        __syncthreads();   // protect buffer s&1 before it is overwritten
    }

#pragma unroll
    for (int tile = 0; tile < 16; ++tile) {
#pragma unroll
        for (int r = 0; r < 8; ++r) {
            const int row = t0 + r + 8 * hi;
            y_v[((size_t)bb * TLEN + row) * CCH + tile * 16 + m] = acc[tile][r];
        }
    }
}

// ---------------------------------------------------------------------------
// Kernel B: stk[row, g] = MLP(dis[row, 0:2]) for row in [0, B*T*KWIN).
// Layer1 (2->64) computed per-lane directly in WMMA A-layout registers.
// Layer2 (64->128) via 128 f32 WMMAs per wave (16 rows).
// Layer3 (128->8) via LDS transpose (stride 132, conflict-free) + 32 WMMAs.
// Block = 4 waves = 64 rows.
// ---------------------------------------------------------------------------
#define H2_STRIDE 132

__global__ __launch_bounds__(128) void mlp_kernel(
    const float* __restrict__ dis,  // (B, T, KWIN, 2) flat rows
    const float* __restrict__ w1,   // (2, 64)
    const float* __restrict__ b1,   // (64,)
    const float* __restrict__ w2,   // (64, 128)
    const float* __restrict__ b2,   // (128,)
    const float* __restrict__ w3,   // (128, 8)
    const float* __restrict__ b3,   // (8,)
    float* __restrict__ stk)        // (B*T*KWIN, 8)
{
    __shared__ float lds_h2[4 * 16 * H2_STRIDE];

    const int lane  = threadIdx.x & 31;
    const int wave  = threadIdx.x >> 5;
    const int rbase = (blockIdx.x * 4 + wave) * 16;
    const int m     = lane & 15;
    const int hi    = lane >> 4;

    // ---- Layer 1: lane l holds h1[row=m][cols ≡ {2hi, 2hi+1} mod 4] ----
    const size_t myrow = (size_t)(rbase + m);
    const float d0 = dis[myrow * 2 + 0];
    const float d1 = dis[myrow * 2 + 1];
    v2f h1[16];
#pragma unroll
    for (int ks = 0; ks < 16; ++ks) {
        const int c0 = 4 * ks + 2 * hi;
        float u0 = b1[c0]     + d0 * w1[c0]     + d1 * w1[64 + c0];
        float u1 = b1[c0 + 1] + d0 * w1[c0 + 1] + d1 * w1[64 + c0 + 1];
        h1[ks].x = u0 > 0.f ? u0 : 0.f;
        h1[ks].y = u1 > 0.f ? u1 : 0.f;
    }

    // ---- Layer 2: D(16x128) = h1(16x64) @ w2(64x128) + b2 ----
    v8f acc[8];
#pragma unroll
    for (int tile = 0; tile < 8; ++tile) {
        acc[tile] = splat8(b2[tile * 16 + m]);
    }
#pragma unroll
    for (int ks = 0; ks < 16; ++ks) {
        const int kk = 4 * ks + 2 * hi;
        const v2f a = h1[ks];
#pragma unroll
        for (int tile = 0; tile < 8; ++tile) {
            v2f bm;
            bm.x = w2[(kk)     * 128 + tile * 16 + m];
            bm.y = w2[(kk + 1) * 128 + tile * 16 + m];
            acc[tile] = wmma_f32_4(a, bm, acc[tile]);
        }
    }

    // ---- ReLU + transpose D-layout -> LDS (row-major, padded stride) ----
    float* hbuf = &lds_h2[wave * 16 * H2_STRIDE];
#pragma unroll
    for (int tile = 0; tile < 8; ++tile) {
#pragma unroll
        for (int r = 0; r < 8; ++r) {
            float v = acc[tile][r];
            v = v > 0.f ? v : 0.f;
            hbuf[(r + 8 * hi) * H2_STRIDE + tile * 16 + m] = v;
        }
    }
    __syncthreads();

    // ---- Layer 3: D(16x16, cols 0..7 valid) = h2(16x128) @ w3(128x8) + b3 --
    v8f acc3 = splat8(m < 8 ? b3[m] : 0.f);
#pragma unroll
    for (int ks = 0; ks < 32; ++ks) {
        const int kk = 4 * ks + 2 * hi;
        v2f a, bm;
        a.x  = hbuf[m * H2_STRIDE + kk];
        a.y  = hbuf[m * H2_STRIDE + kk + 1];
        bm.x = (m < 8) ? w3[(kk)     * 8 + m] : 0.f;
        bm.y = (m < 8) ? w3[(kk + 1) * 8 + m] : 0.f;
        acc3 = wmma_f32_4(a, bm, acc3);
    }

    if (m < 8) {
#pragma unroll
        for (int r = 0; r < 8; ++r) {
            const size_t row = (size_t)(rbase + r + 8 * hi);
            stk[row * 8 + m] = acc3[r];
        }
    }
}

// ---------------------------------------------------------------------------
// Kernel C: mask + softmax over window, weighted sum of y_v (+ bias for OOB).
// One thread per (token, g); out[(b*T+t)*256 + g*32 + d].
// ---------------------------------------------------------------------------
__global__ __launch_bounds__(256) void attn_out_kernel(
    const float* __restrict__ stk,   // (B*T*KWIN, 8)
    const int*   __restrict__ pm,    // (B, T)
    const float* __restrict__ y_v,   // (B, T, C)
    const float* __restrict__ kv_b,  // (2C,)
    float* __restrict__ out)         // (B*T, 256)
{
    const int tid   = blockIdx.x * blockDim.x + threadIdx.x;
    const int g     = tid & 7;
    const int token = tid >> 3;                  // b*T + t
    const int bb    = token >> 13;               // TLEN = 8192
    const int tt    = token & (TLEN - 1);

    const int pmc = pm[token];

    float sv[KWIN];
    float mx = -INFINITY;
#pragma unroll
    for (int k = 0; k < KWIN; ++k) {
        const int p = tt + k - PADW;
        const bool inr = (p >= 0) && (p < TLEN);
        const bool on  = (pmc != 0) && inr && (pm[bb * TLEN + p] != 0);
        const float s  = on ? stk[((size_t)token * KWIN + k) * GH + g] : MASKV;
        sv[k] = s;
        mx = fmaxf(mx, s);
    }

    float attn[KWIN];
    float esum = 0.f;
#pragma unroll
    for (int k = 0; k < KWIN; ++k) {
        attn[k] = __expf(sv[k] - mx);
        esum += attn[k];
    }
    const float inv = 1.f / esum;
    float oob = 0.f;
#pragma unroll
    for (int k = 0; k < KWIN; ++k) {
        attn[k] *= inv;
        const int p = tt + k - PADW;
        if (p < 0 || p >= TLEN) oob += attn[k];
    }

    const int jbase = g * HD;
#pragma unroll
    for (int dc = 0; dc < HD / 4; ++dc) {
        const float4 b4 = *(const float4*)&kv_b[CCH + jbase + dc * 4];
        float4 a4 = make_float4(oob * b4.x, oob * b4.y, oob * b4.z, oob * b4.w);
#pragma unroll
        for (int k = 0; k < KWIN; ++k) {
            const int p = tt + k - PADW;
            if (p >= 0 && p < TLEN) {
                const float4 y4 = *(const float4*)
                    &y_v[((size_t)bb * TLEN + p) * CCH + jbase + dc * 4];
                a4.x += attn[k] * y4.x;
                a4.y += attn[k] * y4.y;
                a4.z += attn[k] * y4.z;
                a4.w += attn[k] * y4.w;
            }
        }
        *(float4*)&out[(size_t)token * CCH + jbase + dc * 4] = a4;
    }
}

// ---------------------------------------------------------------------------
extern "C" void kernel_launch(void* const* d_in, const int* in_sizes, int n_in,
                              void* d_out, int out_size, void* d_ws, size_t ws_size,
                              hipStream_t stream) {
    const float* x    = (const float*)d_in[0];   // (4, 256, 8192)
    const float* dis  = (const float*)d_in[1];   // (4, 8192, 9, 2)
    const int*   pmsk = (const int*)  d_in[2];   // (4, 8192)
    const float* kv_w = (const float*)d_in[3];   // (256, 512)
    const float* kv_b = (const float*)d_in[4];   // (512,)
    const float* w1   = (const float*)d_in[5];
    const float* b1   = (const float*)d_in[6];
    const float* w2   = (const float*)d_in[7];
    const float* b2   = (const float*)d_in[8];
    const float* w3   = (const float*)d_in[9];
    const float* b3   = (const float*)d_in[10];
    float* outp = (float*)d_out;

    // Workspace: y_v (32 MB) | stk (9 MB) | w_il (0.25 MB)
    const size_t yv_bytes  = (size_t)BATCH * TLEN * CCH * sizeof(float);
    const size_t stk_bytes = (size_t)BATCH * TLEN * KWIN * GH * sizeof(float);
    float* y_v  = (float*)d_ws;
    float* stk  = (float*)((char*)d_ws + yv_bytes);
    float* w_il = (float*)((char*)d_ws + yv_bytes + stk_bytes);

    // D: interleave the V-half of kv_w (128 pair-rows x 256 cols)
    interleave_w_kernel<<<128, 256, 0, stream>>>(kv_w, w_il);

    // A: 2048 strips, 8 waves/block -> 256 blocks of 256 threads
    kv_gemm_kernel<<<256, 256, 0, stream>>>(x, w_il, kv_b, y_v);

    // B: 294912 rows, 64 rows/block -> 4608 blocks of 128 threads
    mlp_kernel<<<4608, 128, 0, stream>>>(dis, w1, b1, w2, b2, w3, b3, stk);

    // C: 32768 tokens * 8 groups = 262144 threads -> 1024 blocks of 256
    attn_out_kernel<<<1024, 256, 0, stream>>>(stk, pmsk, y_v, kv_b, outp);
}